// GeneralBranch_24936580120845
// MI455X (gfx1250) — compile-verified
//
#include <hip/hip_runtime.h>

typedef __attribute__((ext_vector_type(2))) float v2f;
typedef __attribute__((ext_vector_type(4))) float v4f;
typedef __attribute__((ext_vector_type(8))) float v8f;

// Problem sizes (fixed by the reference)
constexpr int Dd = 768;          // audio feature dim
constexpr int Tt = 512;          // CLIP text dim
constexpr int Vv = 49408;        // vocab
constexpr int R  = 64 * 8;       // B*N = 512 keyword rows
constexpr int VT = Vv / 16;      // 3088 vocab tiles of 16
constexpr int NP = VT / 2;       // 1544 vocab tile-pairs (32 cols each)
constexpr int S  = 128;          // parallel vocab streams (grid.y * 4 waves)
constexpr float EPS = 1e-8f;

__device__ __forceinline__ v8f wmma_f32_k4(v2f a, v2f b, v8f c) {
  // D = A(16x4) * B(4x16) + C, fp32 native tensor op
  return __builtin_amdgcn_wmma_f32_16x16x4_f32(
      /*neg_a=*/false, a, /*neg_b=*/false, b,
      /*c_mod=*/(short)0, c, /*reuse_a=*/false, /*reuse_b=*/false);
}

__device__ __forceinline__ void upd(float& bs, int& bv, float s, int v) {
  if (s > bs || (s == bs && v < bv)) { bs = s; bv = v; }  // JAX argmax tie: lowest idx
}

__global__ void zero_f32(float* __restrict__ p, int n) {
  int i = blockIdx.x * blockDim.x + threadIdx.x;
  if (i < n) p[i] = 0.0f;
}

// kw = audio @ W_proj + b ; also accumulate per-row sum of squares.
// One wave per 16x16 output tile. grid = (R/16, T/16), block = 32.
__global__ void __launch_bounds__(32)
proj_kernel(const float* __restrict__ audio, const float* __restrict__ W,
            const float* __restrict__ bias, float* __restrict__ kw,
            float* __restrict__ sumsq) {
  const int lane = threadIdx.x;
  const int hi   = lane >> 4;      // lane half selects K pair
  const int l15  = lane & 15;
  const int mt = blockIdx.x, nt = blockIdx.y;
  const int m = mt * 16 + l15;     // A-fragment row this lane feeds
  const int n = nt * 16 + l15;     // B-fragment col this lane feeds

  const float* Arow = audio + (size_t)m * Dd;
  v8f acc = {};
#pragma unroll 4
  for (int k = 0; k < Dd; k += 4) {
    const int klo = k + 2 * hi;
    v2f a = *(const v2f*)(Arow + klo);
    v2f b;
    b.x = W[(size_t)klo * Tt + n];
    b.y = W[(size_t)(klo + 1) * Tt + n];
    acc = wmma_f32_k4(a, b, acc);
  }

  const float bn = bias[n];
#pragma unroll
  for (int i = 0; i < 8; ++i) {
    const int row = mt * 16 + i + 8 * hi;              // C layout: lanes16-31 -> rows+8
    float val = acc[i] + bn;
    kw[(size_t)row * Tt + n] = val;
    float s = val * val;
    s += __shfl_xor(s, 1);
    s += __shfl_xor(s, 2);
    s += __shfl_xor(s, 4);
    s += __shfl_xor(s, 8);                             // reduce over 16 cols of this half
    if (l15 == 0) atomicAdd(&sumsq[row], s);
  }
}

// emb_norm[v] = ||token_emb[v,:]||, one wave per vocab row.
__global__ void __launch_bounds__(256)
embnorm_kernel(const float* __restrict__ emb, float* __restrict__ en) {
  const int wave = threadIdx.x >> 5, lane = threadIdx.x & 31;
  const int v = blockIdx.x * 8 + wave;
  const float* row = emb + (size_t)v * Tt;
  float s = 0.0f;
#pragma unroll
  for (int j = 0; j < 4; ++j) {
    v4f d = *(const v4f*)(row + (j * 32 + lane) * 4);
    s += d.x * d.x + d.y * d.y + d.z * d.z + d.w * d.w;
  }
  s += __shfl_xor(s, 1);  s += __shfl_xor(s, 2);  s += __shfl_xor(s, 4);
  s += __shfl_xor(s, 8);  s += __shfl_xor(s, 16);
  if (lane == 0) en[v] = sqrtf(s);
}

// Fused cosine-score GEMM + running argmax, 2x2 register-blocked.
// grid = (R/32, 32), block = 128 (4 waves share a 32-row LDS A tile;
// each wave owns an independent vocab stream of 32-column tile pairs).
__global__ void __launch_bounds__(128)
score_kernel(const float* __restrict__ kw, const float* __restrict__ sumsq,
             const float* __restrict__ emb, const float* __restrict__ en,
             float* __restrict__ pscore, int* __restrict__ pidx) {
  // 32 rows x 512 cols, XOR-swizzled (c ^ 4*(r&15)) for conflict-free ds_load_b64
  __shared__ float Atile[32 * Tt];  // exactly 64 KB
  const int tid  = threadIdx.x;
  const int wave = tid >> 5, lane = tid & 31;
  const int hi = lane >> 4, l15 = lane & 15;
  const int rst    = blockIdx.x;             // row super-tile (32 rows)
  const int stream = blockIdx.y * 4 + wave;  // 0..S-1

  // stage kw[rst*32 .. +32][0..T) into LDS with XOR swizzle
  for (int r = wave; r < 32; r += 4) {
    const float* src = kw + (size_t)(rst * 32 + r) * Tt;
    const int swz = 4 * (r & 15);
#pragma unroll
    for (int j = 0; j < 4; ++j) {
      const int c = j * 128 + lane * 4;
      *(v4f*)(Atile + r * Tt + (c ^ swz)) = *(const v4f*)(src + c);
    }
  }
  __syncthreads();

  const int aswz = 4 * l15;                  // per-lane LDS swizzle for A fragments
  float rnorm[2][8];
#pragma unroll
  for (int r = 0; r < 2; ++r)
#pragma unroll
    for (int i = 0; i < 8; ++i)
      rnorm[r][i] = sqrtf(sumsq[rst * 32 + r * 16 + i + 8 * hi]);

  float best[2][8]; int bidx[2][8];
#pragma unroll
  for (int r = 0; r < 2; ++r)
#pragma unroll
    for (int i = 0; i < 8; ++i) { best[r][i] = -3.4e38f; bidx[r][i] = 0x7fffffff; }

  for (int p = stream; p < NP; p += S) {
    const int v0 = (2 * p) * 16 + l15;       // this lane's two vocab columns
    const int v1 = (2 * p + 1) * 16 + l15;
    if (p + S < NP) {                        // prefetch next pair (global_prefetch_b8)
      __builtin_prefetch(emb + (size_t)(2 * (p + S)) * 16 * Tt + (size_t)l15 * Tt, 0, 1);
      __builtin_prefetch(emb + (size_t)(2 * (p + S) + 1) * 16 * Tt + (size_t)l15 * Tt, 0, 1);
    }
    const float en0 = en[v0], en1 = en[v1];
    const float* B0 = emb + (size_t)v0 * Tt;
    const float* B1 = emb + (size_t)v1 * Tt;

    v8f a00 = {}, a01 = {}, a10 = {}, a11 = {};
#pragma unroll 4
    for (int k = 0; k < Tt; k += 4) {
      const int klo = k + 2 * hi;
      v2f fa0 = *(const v2f*)(Atile + l15 * Tt + (klo ^ aswz));         // ds_load_b64
      v2f fa1 = *(const v2f*)(Atile + (16 + l15) * Tt + (klo ^ aswz));
      v2f fb0 = *(const v2f*)(B0 + klo);                                // global_load_b64
      v2f fb1 = *(const v2f*)(B1 + klo);
      a00 = wmma_f32_k4(fa0, fb0, a00);
      a01 = wmma_f32_k4(fa0, fb1, a01);
      a10 = wmma_f32_k4(fa1, fb0, a10);
      a11 = wmma_f32_k4(fa1, fb1, a11);
    }

#pragma unroll
    for (int i = 0; i < 8; ++i) {
      upd(best[0][i], bidx[0][i], a00[i] / fmaxf(rnorm[0][i] * en0, EPS), v0);
      upd(best[0][i], bidx[0][i], a01[i] / fmaxf(rnorm[0][i] * en1, EPS), v1);
      upd(best[1][i], bidx[1][i], a10[i] / fmaxf(rnorm[1][i] * en0, EPS), v0);
      upd(best[1][i], bidx[1][i], a11[i] / fmaxf(rnorm[1][i] * en1, EPS), v1);
    }
  }

  // argmax across the 16 columns held by this lane half
#pragma unroll
  for (int r = 0; r < 2; ++r)
#pragma unroll
    for (int i = 0; i < 8; ++i) {
      float s = best[r][i]; int v = bidx[r][i];
#pragma unroll
      for (int mask = 1; mask <= 8; mask <<= 1) {
        const float so = __shfl_xor(s, mask);
        const int vo = __shfl_xor(v, mask);
        if (so > s || (so == s && vo < v)) { s = so; v = vo; }
      }
      if (l15 == 0) {
        const int row = rst * 32 + r * 16 + i + 8 * hi;
        pscore[(size_t)row * S + stream] = s;
        pidx[(size_t)row * S + stream]   = v;
      }
    }
}

// Reduce stream partials per row; gather winning embedding row to output.
__global__ void __launch_bounds__(256)
finalize_kernel(const float* __restrict__ pscore, const int* __restrict__ pidx,
                const float* __restrict__ emb, float* __restrict__ out) {
  const int wave = threadIdx.x >> 5, lane = threadIdx.x & 31;
  const int row = blockIdx.x * 8 + wave;
  float s = -3.4e38f; int v = 0x7fffffff;
#pragma unroll
  for (int j = 0; j < 4; ++j) {
    const float sj = pscore[(size_t)row * S + j * 32 + lane];
    const int   vj = pidx[(size_t)row * S + j * 32 + lane];
    if (sj > s || (sj == s && vj < v)) { s = sj; v = vj; }
  }
#pragma unroll
  for (int mask = 1; mask <= 16; mask <<= 1) {
    const float so = __shfl_xor(s, mask);
    const int vo = __shfl_xor(v, mask);
    if (so > s || (so == s && vo < v)) { s = so; v = vo; }
  }
  const float* src = emb + (size_t)v * Tt;
  float* dst = out + (size_t)row * Tt;
#pragma unroll
  for (int j = 0; j < 4; ++j) {
    const int c = (j * 32 + lane) * 4;
    *(v4f*)(dst + c) = *(const v4f*)(src + c);
  }
}

extern "C" void kernel_launch(void* const* d_in, const int* in_sizes, int n_in,
                              void* d_out, int out_size, void* d_ws, size_t ws_size,
                              hipStream_t stream) {
  const float* audio = (const float*)d_in[0];  // [64,8,768]
  const float* W     = (const float*)d_in[1];  // [768,512]
  const float* bias  = (const float*)d_in[2];  // [512]
  const float* emb   = (const float*)d_in[3];  // [49408,512]
  float* out = (float*)d_out;                  // [64,8,512]

  float* ws    = (float*)d_ws;
  float* kw    = ws;                           // R*T   = 262144 f
  float* sumsq = kw + (size_t)R * Tt;          // R     = 512 f
  float* en    = sumsq + R;                    // V     = 49408 f
  float* psc   = en + Vv;                      // R*S   = 65536 f
  int*   pix   = (int*)(psc + (size_t)R * S);  // R*S   = 65536 i

  zero_f32<<<(R + 255) / 256, 256, 0, stream>>>(sumsq, R);
  proj_kernel<<<dim3(R / 16, Tt / 16), 32, 0, stream>>>(audio, W, bias, kw, sumsq);
  embnorm_kernel<<<Vv / 8, 256, 0, stream>>>(emb, en);
  score_kernel<<<dim3(R / 32, S / 4), 128, 0, stream>>>(kw, sumsq, emb, en, psc, pix);
  finalize_kernel<<<R / 8, 256, 0, stream>>>(psc, pix, emb, out);
}